// ExpSmooth_66718021976484
// MI455X (gfx1250) — compile-verified
//
#include <hip/hip_runtime.h>
#include <hip/hip_bf16.h>

#define N_SERIES 4096
#define T_STEPS  1024
#define CONDN    896
#define PREDN    56
#define PERIODN  7
#define NJ       129   // T - COND + 1
#define NSW      135   // NJ + PERIOD - 1

typedef __attribute__((ext_vector_type(2))) float v2f;
typedef __attribute__((ext_vector_type(8))) float v8f;

__device__ __forceinline__ float sigmoidf(float x) {
    return 1.0f / (1.0f + __expf(-x));
}

// ---------------------------------------------------------------------------
// Kernel 1: sequential Holt-Winters scan, one thread per series.
// Seasonal ring buffer in named registers (unroll-by-7); the t-loop is split
// at the phase-aligned boundaries t=889 (=127*7) and t=896 (=128*7) so the
// hot 127-block loop has zero conditionals.
// ---------------------------------------------------------------------------
__global__ __launch_bounds__(128) void hw_scan_kernel(
    const float* __restrict__ query, const float* __restrict__ params,
    float* __restrict__ sm, float* __restrict__ diffs,
    float* __restrict__ lastL, float* __restrict__ lastT,
    float* __restrict__ seasW)
{
    int n = blockIdx.x * blockDim.x + threadIdx.x;
    const float* q     = query + (size_t)n * T_STEPS;
    float*       smrow = sm    + (size_t)n * T_STEPS;
    const float* p     = params + n * (5 + PERIODN);

    float L  = p[0];
    float a  = sigmoidf(p[1]);
    float Tr = p[2];
    float b  = sigmoidf(p[3]);
    float g  = p[4];                       // season_sm: NOT sigmoided in reference
    float s0 = sigmoidf(p[5]),  s1 = sigmoidf(p[6]),  s2 = sigmoidf(p[7]);
    float s3 = sigmoidf(p[8]),  s4 = sigmoidf(p[9]),  s5 = sigmoidf(p[10]);
    float s6 = sigmoidf(p[11]);
    float acc = 0.0f;

    float* ll  = lastL + n * NJ;
    float* lt  = lastT + n * NJ;
    float* swp = seasW + n * NSW;

    int t = 0;
#define HW_STEP(SREG, DOSEAS, DOLVL) {                                      \
        float y    = q[t];                                                  \
        float pred = L + Tr;                                                \
        float e    = y - pred;                                              \
        float lv   = fmaf(a, e, pred);            /* a*y + (1-a)*pred   */  \
        acc += fabsf(lv - L);                                               \
        float ntr  = fmaf(b, (lv - L) - Tr, Tr);  /* b*dL + (1-b)*Tr    */  \
        float ns   = fmaf(g, e - (SREG), (SREG)); /* g*(y-pred)+(1-g)s  */  \
        __builtin_nontemporal_store(y - lv - ns, &smrow[t]);                \
        if (DOLVL) {                                                        \
            ll[t - (CONDN - 1)] = lv;                                       \
            lt[t - (CONDN - 1)] = ntr;                                      \
        }                                                                   \
        if (DOSEAS) { swp[t - (CONDN - PERIODN)] = ns; }                    \
        L = lv; Tr = ntr; (SREG) = ns;                                      \
        ++t; }

    // Phase A: t = 0..888  (127 blocks of 7, no tail outputs)
    for (int blk = 0; blk < 127; ++blk) {
        __builtin_prefetch(q + t + 28, 0, 0);   // gfx1250 global_prefetch_b8
        HW_STEP(s0,0,0) HW_STEP(s1,0,0) HW_STEP(s2,0,0) HW_STEP(s3,0,0)
        HW_STEP(s4,0,0) HW_STEP(s5,0,0) HW_STEP(s6,0,0)
    }
    // t = 889..894: seasonal window only (ring phase s0..s5)
    HW_STEP(s0,1,0) HW_STEP(s1,1,0) HW_STEP(s2,1,0)
    HW_STEP(s3,1,0) HW_STEP(s4,1,0) HW_STEP(s5,1,0)
    // t = 895: first level/trend tap (phase s6)
    HW_STEP(s6,1,1)
    // t = 896..1021: 18 blocks of 7, all tail outputs (phase restarts at s0)
    for (int blk = 0; blk < 18; ++blk) {
        __builtin_prefetch(q + t + 28, 0, 0);
        HW_STEP(s0,1,1) HW_STEP(s1,1,1) HW_STEP(s2,1,1) HW_STEP(s3,1,1)
        HW_STEP(s4,1,1) HW_STEP(s5,1,1) HW_STEP(s6,1,1)
    }
    // t = 1022, 1023 (phase s0, s1)
    HW_STEP(s0,1,1) HW_STEP(s1,1,1)
#undef HW_STEP

    diffs[n] = acc * (1.0f / (float)T_STEPS);
}

// ---------------------------------------------------------------------------
// Kernel 2: pr[n,j,h] = L[n,j] + Tr[n,j]*(h+1) + seasW[n, j + h%7]
// Exact rank-9 f32 GEMM per series:  pr = A(129x9,pad12) x B(12x56)
//   B row0 = ones, row1 = horizon, rows2..8 = one-hot(h mod 7)
// 3 chained V_WMMA_F32_16X16X4_F32 per 16x16 tile; one wave per (n, j-tile).
// Full tiles (jt<8) store unguarded with immediate offsets off one base addr.
// ---------------------------------------------------------------------------
__global__ __launch_bounds__(256) void hw_pr_wmma_kernel(
    const float* __restrict__ lastL, const float* __restrict__ lastT,
    const float* __restrict__ seasW, float* __restrict__ pr)
{
    int wave = (blockIdx.x * blockDim.x + threadIdx.x) >> 5;
    int lane = threadIdx.x & 31;
    int n  = wave / 9;          // series
    int jt = wave % 9;          // j-tile (9 tiles of 16 cover 129 rows)
    int j0 = jt * 16;
    int m  = lane & 15;         // A row within tile / D column
    int hi = lane >> 4;         // lane half selects K pair / D row half

    int jc = j0 + m; if (jc > NJ - 1) jc = NJ - 1;   // clamp pad rows (never stored)
    float Lv = lastL[n * NJ + jc];
    float Tv = lastT[n * NJ + jc];
    const float* sw = seasW + n * NSW + jc;
    float S0 = sw[0], S1 = sw[1], S2 = sw[2], S3 = sw[3];
    float S4 = sw[4], S5 = sw[5], S6 = sw[6];

    // A fragments: lanes<16 hold K={4c,4c+1}, lanes>=16 hold K={4c+2,4c+3}
    // K order: [L, Tr, S0..S6, 0, 0, 0]
    v2f a0, a1, a2;
    a0.x = hi ? S0 : Lv;    a0.y = hi ? S1 : Tv;
    a1.x = hi ? S4 : S2;    a1.y = hi ? S5 : S3;
    a2.x = hi ? 0.0f : S6;  a2.y = 0.0f;

    // Single base address; every store below is an immediate offset off it.
    // Element (row = j0 + 8*hi + v, col = ht*16 + m) -> basep[ht*16 + v*56]
    float* basep = pr + (size_t)n * NJ * PREDN
                      + (size_t)(j0 + 8 * hi) * PREDN + m;
    bool full_rows = (jt < 8);   // wave-uniform

#pragma unroll
    for (int ht = 0; ht < 4; ++ht) {            // 4 h-tiles cover 56 cols
        int h = ht * 16 + m;
        int r = h % 7;
        v2f b0, b1, b2;
        b0.x = hi ? (r == 0 ? 1.0f : 0.0f) : 1.0f;              // K=2 | K=0
        b0.y = hi ? (r == 1 ? 1.0f : 0.0f) : (float)(h + 1);    // K=3 | K=1
        b1.x = (r == (hi ? 4 : 2)) ? 1.0f : 0.0f;               // K=6 | K=4
        b1.y = (r == (hi ? 5 : 3)) ? 1.0f : 0.0f;               // K=7 | K=5
        b2.x = hi ? 0.0f : (r == 6 ? 1.0f : 0.0f);              // K=10| K=8
        b2.y = 0.0f;                                            // K=11| K=9

        v8f c = {};
        c = __builtin_amdgcn_wmma_f32_16x16x4_f32(false, a0, false, b0,
                                                  (short)0, c, false, false);
        c = __builtin_amdgcn_wmma_f32_16x16x4_f32(false, a1, false, b1,
                                                  (short)0, c, false, false);
        c = __builtin_amdgcn_wmma_f32_16x16x4_f32(false, a2, false, b2,
                                                  (short)0, c, false, false);

        if (full_rows) {
            if (ht < 3) {                        // compile-time: all 16 cols valid
#pragma unroll
                for (int v = 0; v < 8; ++v)
                    __builtin_nontemporal_store(c[v], &basep[ht * 16 + v * PREDN]);
            } else {                             // last h-tile: cols 48..55 only
                if (m < 8) {
#pragma unroll
                    for (int v = 0; v < 8; ++v)
                        __builtin_nontemporal_store(c[v], &basep[ht * 16 + v * PREDN]);
                }
            }
        } else {
            // jt == 8: only row 128 exists -> v == 0 on the low lane half
            if (hi == 0 && (ht < 3 || m < 8))
                __builtin_nontemporal_store(c[0], &basep[ht * 16]);
        }
    }
}

extern "C" void kernel_launch(void* const* d_in, const int* in_sizes, int n_in,
                              void* d_out, int out_size, void* d_ws, size_t ws_size,
                              hipStream_t stream) {
    (void)in_sizes; (void)n_in; (void)out_size; (void)ws_size;
    const float* query  = (const float*)d_in[0];
    const float* params = (const float*)d_in[1];

    float* out   = (float*)d_out;
    float* sm    = out;                                        // N*T
    float* pr    = out + (size_t)N_SERIES * T_STEPS;           // N*129*56
    float* diffs = pr  + (size_t)N_SERIES * NJ * PREDN;        // N

    float* ws    = (float*)d_ws;
    float* lastL = ws;                                         // N*129
    float* lastT = lastL + (size_t)N_SERIES * NJ;              // N*129
    float* seasW = lastT + (size_t)N_SERIES * NJ;              // N*135

    hw_scan_kernel<<<N_SERIES / 128, 128, 0, stream>>>(
        query, params, sm, diffs, lastL, lastT, seasW);

    // 4096 series * 9 j-tiles = 36864 waves; 8 waves per 256-thread block
    hw_pr_wmma_kernel<<<(N_SERIES * 9) / 8, 256, 0, stream>>>(
        lastL, lastT, seasW, pr);
}